// Detection_84894323572772
// MI455X (gfx1250) — compile-verified
//
#include <hip/hip_runtime.h>
#include <cstdint>

// ---------------------------------------------------------------------------
// Problem constants (fixed by the reference: B=16, C=64, H=W=256, k=512)
// ---------------------------------------------------------------------------
#define BATCH 16
#define CHANS 64
#define HDIM 256
#define WDIM 256
#define HW   (HDIM * WDIM)        // 65536
#define KSEL 512
#define FEAT_ELEMS (BATCH * CHANS * KSEL)   // 524288 floats, then location follows

#define NTHREADS 1024
#define NBINS 4096                 // top-12 bits of monotonic key
#define CAND_CAP 2048              // sorted capacity (power of two for bitonic)

// Types matching the builtin's expected parameters:
//   param0: AS1 (global)  pointer to int __attribute__((vector_size(16)))
//   param1: AS3 (local)   pointer to same
typedef int v4i_ __attribute__((vector_size(4 * sizeof(int))));
typedef __attribute__((address_space(1))) v4i_* g_v4i_ptr;
typedef __attribute__((address_space(3))) v4i_* l_v4i_ptr;

// ---------------------------------------------------------------------------
// gfx1250 async global->LDS DMA (ASYNCcnt path). Builtin if present, else asm.
// ---------------------------------------------------------------------------
__device__ __forceinline__ void async_copy_b128(const void* gsrc, void* ldst) {
#if __has_builtin(__builtin_amdgcn_global_load_async_to_lds_b128)
  __builtin_amdgcn_global_load_async_to_lds_b128(
      (g_v4i_ptr)(uintptr_t)gsrc,
      (l_v4i_ptr)(uint32_t)(uintptr_t)ldst,
      /*offset=*/0, /*cpol=*/0);
#else
  asm volatile("global_load_async_to_lds_b128 %0, %1, off"
               :: "v"((uint32_t)(uintptr_t)ldst),
                  "v"((uint64_t)(uintptr_t)gsrc)
               : "memory");
#endif
}

__device__ __forceinline__ void wait_async0() {
#if __has_builtin(__builtin_amdgcn_s_wait_asynccnt)
  __builtin_amdgcn_s_wait_asynccnt(0);
#else
  asm volatile("s_wait_asynccnt 0" ::: "memory");
#endif
}

// Order-preserving float -> uint32 key (larger key <=> larger float).
__device__ __forceinline__ uint32_t mono32(float f) {
  uint32_t u = __float_as_uint(f);
  return (u & 0x80000000u) ? ~u : (u | 0x80000000u);
}

// ---------------------------------------------------------------------------
// Kernel 1: per-batch top-512 selection, fully resident in LDS.
// grid = 16 blocks (one per batch), block = 1024 threads (32 waves).
// Writes: idxws[b*512 + j] (descending-value order, JAX tie-break) and the
// location part of d_out. LDS budget ~292 KB < 320 KB/WG (CDNA5).
// ---------------------------------------------------------------------------
__global__ void __launch_bounds__(NTHREADS)
topk_select_kernel(const float* __restrict__ input,
                   float* __restrict__ out,
                   int* __restrict__ idxws) {
  __shared__ __align__(16) float heat_s[HW];              // 256 KB
  __shared__ uint32_t hist[NBINS];                        // 16 KB
  __shared__ unsigned long long cand[CAND_CAP];           // 16 KB
  __shared__ uint32_t suf[NTHREADS];                      // 4 KB
  __shared__ uint32_t thrBin;
  __shared__ uint32_t candCount;

  const int b = blockIdx.x;
  const int tid = threadIdx.x;
  const float* heat = input + (size_t)b * CHANS * HW;     // channel 0 plane

  // ---- Phase 0: async-DMA the whole heatmap into LDS (one HBM pass) ----
  #pragma unroll
  for (int i = 0; i < 16; ++i) {
    const int elem = (tid + i * NTHREADS) * 4;            // 4 floats per b128
    async_copy_b128(heat + elem, &heat_s[elem]);
  }
  // Zero histogram while DMA is in flight.
  #pragma unroll
  for (int i = 0; i < NBINS / NTHREADS; ++i) hist[tid + i * NTHREADS] = 0u;
  if (tid == 0) candCount = 0u;
  wait_async0();
  __syncthreads();

  // ---- Phase 1: 4096-bin histogram of top-12 key bits ----
  #pragma unroll 4
  for (int i = 0; i < HW / NTHREADS; ++i) {
    const int idx = tid + (i << 10);
    atomicAdd(&hist[mono32(heat_s[idx]) >> 20], 1u);
  }
  __syncthreads();

  // ---- Phase 2: suffix scan to locate the threshold bin ----
  const int binBase = tid * 4;
  uint32_t own = hist[binBase] + hist[binBase + 1] + hist[binBase + 2] + hist[binBase + 3];
  suf[tid] = own;
  __syncthreads();
  for (int d = 1; d < NTHREADS; d <<= 1) {
    uint32_t v = (tid + d < NTHREADS) ? suf[tid + d] : 0u;
    __syncthreads();
    suf[tid] += v;
    __syncthreads();
  }
  {
    uint32_t run = suf[tid] - own;        // elements in bins above this thread's range
    #pragma unroll
    for (int j = 3; j >= 0; --j) {
      const uint32_t h = hist[binBase + j];
      if (run < KSEL && run + h >= KSEL) thrBin = (uint32_t)(binBase + j);
      run += h;
    }
  }
  __syncthreads();
  const uint32_t T = thrBin;

  // ---- Phase 3: compact candidates (bin >= T) as 48-bit composites ----
  // composite = (monokey << 16) | (0xFFFF - idx): descending sort == value
  // descending with lower-index-first tie break (JAX top_k semantics).
  #pragma unroll 4
  for (int i = 0; i < HW / NTHREADS; ++i) {
    const uint32_t idx = (uint32_t)(tid + (i << 10));
    const uint32_t m = mono32(heat_s[idx]);
    if ((m >> 20) >= T) {
      const uint32_t pos = atomicAdd(&candCount, 1u);
      if (pos < CAND_CAP)
        cand[pos] = ((unsigned long long)m << 16) | (unsigned long long)(0xFFFFu - idx);
    }
  }
  __syncthreads();
  const uint32_t P = (candCount < CAND_CAP) ? candCount : CAND_CAP;
  for (uint32_t t = tid; t < CAND_CAP; t += NTHREADS)
    if (t >= P) cand[t] = 0ull;           // pad sorts to the tail
  // ---- Phase 4: bitonic sort, descending ----
  for (uint32_t kk = 2; kk <= CAND_CAP; kk <<= 1) {
    for (uint32_t jj = kk >> 1; jj > 0; jj >>= 1) {
      __syncthreads();
      #pragma unroll
      for (uint32_t t = tid; t < CAND_CAP; t += NTHREADS) {
        const uint32_t ixj = t ^ jj;
        if (ixj > t) {
          const unsigned long long a = cand[t];
          const unsigned long long c = cand[ixj];
          const bool descBlock = ((t & kk) == 0);
          if (descBlock ? (a < c) : (a > c)) { cand[t] = c; cand[ixj] = a; }
        }
      }
    }
  }
  __syncthreads();

  // ---- Phase 5: emit indices + location ----
  if (tid < KSEL) {
    const unsigned long long cpq = cand[tid];
    const uint32_t idx = 0xFFFFu - (uint32_t)(cpq & 0xFFFFull);
    idxws[(b << 9) + tid] = (int)idx;
    const uint32_t col = idx & (WDIM - 1);
    const uint32_t row = idx >> 8;
    // xs = linspace(-1,1,256); ys = linspace(1,-1,256)
    const float x = (float)col * (2.0f / 255.0f) - 1.0f;
    const float y = 1.0f - (float)row * (2.0f / 255.0f);
    float* loc = out + FEAT_ELEMS + b * (2 * KSEL);
    loc[tid] = x;
    loc[KSEL + tid] = y;
  }
}

// ---------------------------------------------------------------------------
// Kernel 2: feature gather. One block per (b,c) plane: scattered 4B loads
// (hot lines live in the 192 MB L2), fully coalesced stores.
// grid = B*C = 1024 blocks, block = 512 threads.
// ---------------------------------------------------------------------------
__global__ void __launch_bounds__(KSEL)
feature_gather_kernel(const float* __restrict__ input,
                      const int* __restrict__ idxws,
                      float* __restrict__ out) {
  const int bc = blockIdx.x;               // b*64 + c
  const int b = bc >> 6;
  const int j = threadIdx.x;               // 0..511
  const int idx = idxws[(b << 9) + j];
  out[((size_t)bc << 9) + j] = input[((size_t)bc << 16) + (uint32_t)idx];
}

// ---------------------------------------------------------------------------
extern "C" void kernel_launch(void* const* d_in, const int* in_sizes, int n_in,
                              void* d_out, int out_size, void* d_ws, size_t ws_size,
                              hipStream_t stream) {
  const float* input = (const float*)d_in[0];  // (16,64,256,256) f32
  // d_in[1] is k == 512 (compile-time constant here)
  float* out = (float*)d_out;                  // features(16,64,512) ++ location(16,2,512)
  int* idxws = (int*)d_ws;                     // 16*512 ints of scratch

  topk_select_kernel<<<dim3(BATCH), dim3(NTHREADS), 0, stream>>>(input, out, idxws);
  feature_gather_kernel<<<dim3(BATCH * CHANS), dim3(KSEL), 0, stream>>>(input, idxws, out);
}